// TemporalDeformableDecoderLayer_75806172774976
// MI455X (gfx1250) — compile-verified
//
#include <hip/hip_runtime.h>
#include <math.h>

// ---- problem constants (match reference setup_inputs) ----
#define Bb   4
#define Qq   300
#define WL   8
#define HH   64
#define WW   64
#define DD   256
#define MM   4
#define PE_D 64
#define BQ   (Bb*Qq)          // 1200

// ---- workspace layout (floats) ----
#define WS_QN   0                         // [BQ,256]
#define WS_REF  (WS_QN  + BQ*DD)          // [BQ,2]
#define WS_PRM  (WS_REF + BQ*2)           // [BQ,Wl,M,6] = cx,cy,alpha,w,t0,t1
#define WS_AGG  (WS_PRM + BQ*WL*MM*6)     // [BQ,256]
#define WS_H    (WS_AGG + BQ*DD)          // [BQ,256]
#define WS_MID  (WS_H   + BQ*DD)          // [BQ,1024]

typedef float v2f __attribute__((ext_vector_type(2)));
typedef float v8f __attribute__((ext_vector_type(8)));

// ---------------------------------------------------------------------------
// LayerNorm: one block (256 threads) per row of [BQ,256]
// ---------------------------------------------------------------------------
__global__ void ln_kernel(const float* __restrict__ in,
                          const float* __restrict__ g,
                          const float* __restrict__ b,
                          float* __restrict__ out) {
    __shared__ float red[DD];
    const int row = blockIdx.x, t = threadIdx.x;
    float x = in[row*DD + t];
    red[t] = x; __syncthreads();
    for (int s = DD/2; s > 0; s >>= 1) { if (t < s) red[t] += red[t+s]; __syncthreads(); }
    float mean = red[0] * (1.0f/DD);
    __syncthreads();
    float dx = x - mean;
    red[t] = dx*dx; __syncthreads();
    for (int s = DD/2; s > 0; s >>= 1) { if (t < s) red[t] += red[t+s]; __syncthreads(); }
    float var = red[0] * (1.0f/DD);
    out[row*DD + t] = dx * rsqrtf(var + 1e-5f) * g[t] + b[t];
}

// ---------------------------------------------------------------------------
// ref = sigmoid(qn @ w_ref + b_ref): one block per row
// ---------------------------------------------------------------------------
__global__ void ref_kernel(const float* __restrict__ qn,
                           const float* __restrict__ w_ref,  // [256,2]
                           const float* __restrict__ b_ref,
                           float* __restrict__ refo) {       // [BQ,2]
    __shared__ float r0[DD], r1[DD];
    const int row = blockIdx.x, t = threadIdx.x;
    float x = qn[row*DD + t];
    r0[t] = x * w_ref[t*2 + 0];
    r1[t] = x * w_ref[t*2 + 1];
    __syncthreads();
    for (int s = DD/2; s > 0; s >>= 1) {
        if (t < s) { r0[t] += r0[t+s]; r1[t] += r1[t+s]; }
        __syncthreads();
    }
    if (t == 0) {
        float a = r0[0] + b_ref[0], c = r1[0] + b_ref[1];
        refo[row*2+0] = 1.0f / (1.0f + expf(-a));
        refo[row*2+1] = 1.0f / (1.0f + expf(-c));
    }
}

// ---------------------------------------------------------------------------
// Per (b,q,wl): PE -> rel -> psi -> {delta(12), logits(4)} -> sampling params
// grid = BQ*Wl blocks, 64 threads
// ---------------------------------------------------------------------------
__global__ void prep_kernel(const float* __restrict__ qn,
                            const float* __restrict__ delta_sec, // [BQ,Wl]
                            const float* __restrict__ w_rel,     // [64,256]
                            const float* __restrict__ b_rel,     // [256]
                            const float* __restrict__ w_delta,   // [512,12]
                            const float* __restrict__ b_delta,
                            const float* __restrict__ w_alpha,   // [512,4]
                            const float* __restrict__ b_alpha,
                            const float* __restrict__ refp,      // [BQ,2]
                            const float* __restrict__ lam,
                            float* __restrict__ prm) {           // [BQ*Wl*M*6]
    __shared__ float pe[PE_D];
    __shared__ float psi[2*DD];
    __shared__ float outs[16];
    const int idx = blockIdx.x;
    const int wl = idx & 7, bq = idx >> 3;
    const int t = threadIdx.x;
    const float ds = delta_sec[bq*WL + wl];

    if (t < PE_D) {
        int i = (t < 32) ? t : (t - 32);
        float freq = expf(-logf(10000.0f) * (float)i / 32.0f);
        float ang = ds * freq;
        pe[t] = (t < 32) ? sinf(ang) : cosf(ang);
    }
    for (int j = t; j < DD; j += 64) psi[j] = qn[bq*DD + j];
    __syncthreads();
    for (int j = t; j < DD; j += 64) {
        float s = b_rel[j];
        #pragma unroll 8
        for (int i = 0; i < PE_D; ++i) s += pe[i] * w_rel[i*DD + j];
        psi[DD + j] = s;
    }
    __syncthreads();
    if (t < 16) {
        float s;
        if (t < 12) {
            s = b_delta[t];
            for (int i = 0; i < 2*DD; ++i) s += psi[i] * w_delta[i*12 + t];
        } else {
            int c = t - 12;
            s = b_alpha[c];
            for (int i = 0; i < 2*DD; ++i) s += psi[i] * w_alpha[i*4 + c];
        }
        outs[t] = s;
    }
    __syncthreads();
    if (t == 0) {
        float lamv = lam[0];
        float lam_sp = (lamv > 20.0f) ? lamv : log1pf(expf(lamv));
        float rx = refp[bq*2+0], ry = refp[bq*2+1];
        float lg[MM], cx[MM], cy[MM], al[MM], t0[MM], t1[MM];
        float mx = -1e30f;
        for (int m = 0; m < MM; ++m) {
            float d0 = outs[m*3+0], d1 = outs[m*3+1], d2 = outs[m*3+2];
            cx[m] = fminf(fmaxf(rx + tanhf(d0), 0.0f), 1.0f);
            cy[m] = fminf(fmaxf(ry + tanhf(d1), 0.0f), 1.0f);
            float dt = tanhf(d2);
            float target = (float)wl + dt;
            float t0f = fminf(fmaxf(floorf(target), 0.0f), (float)(WL-1));
            t0[m] = t0f;
            t1[m] = fminf(t0f + 1.0f, (float)(WL-1));
            al[m] = target - t0f;
            lg[m] = outs[12+m] - lam_sp * fabsf(ds);
            mx = fmaxf(mx, lg[m]);
        }
        float den = 0.0f, ex[MM];
        for (int m = 0; m < MM; ++m) { ex[m] = expf(lg[m] - mx); den += ex[m]; }
        float inv = 1.0f / den;
        for (int m = 0; m < MM; ++m) {
            float* p = prm + (size_t)(idx*MM + m)*6;
            p[0] = cx[m]; p[1] = cy[m]; p[2] = al[m];
            p[3] = ex[m]*inv; p[4] = t0[m]; p[5] = t1[m];
        }
    }
}

// ---------------------------------------------------------------------------
// Deformable sampling + aggregation. One block per (b,q), 64 threads,
// each thread owns 4 channels (float4 b128 loads, fully coalesced, L2-resident)
// ---------------------------------------------------------------------------
__global__ void sample_kernel(const float* __restrict__ feats,      // [B,Wl,H,W,256]
                              const unsigned char* __restrict__ msk,// [B,Wl,H,W]
                              const float* __restrict__ prm,        // [BQ*32*6]
                              float* __restrict__ agg) {            // [BQ,256]
    __shared__ float P[WL*MM][6];
    const int bq = blockIdx.x, t = threadIdx.x;
    const int bb = bq / Qq;
    if (t < WL*MM) {
        const float* p = prm + (size_t)(bq*WL*MM + t)*6;
        #pragma unroll
        for (int j = 0; j < 6; ++j) P[t][j] = p[j];
    }
    __syncthreads();
    const float4* f4 = reinterpret_cast<const float4*>(feats);
    float a0 = 0.f, a1 = 0.f, a2 = 0.f, a3 = 0.f;
    for (int p = 0; p < WL*MM; ++p) {
        const float cx = P[p][0], cy = P[p][1], al = P[p][2], wgt = P[p][3];
        const int tt0 = (int)P[p][4], tt1 = (int)P[p][5];
        const float px = cx * WW - 0.5f, py = cy * HH - 0.5f;
        const float x0f = floorf(px), y0f = floorf(py);
        const float wx = px - x0f, wy = py - y0f;
        int x0i = (int)x0f, y0i = (int)y0f;
        int x1i = min(max(x0i + 1, 0), WW-1);
        int y1i = min(max(y0i + 1, 0), HH-1);
        x0i = min(max(x0i, 0), WW-1);
        y0i = min(max(y0i, 0), HH-1);
        const float w00 = (1.f-wy)*(1.f-wx), w01 = (1.f-wy)*wx;
        const float w10 = wy*(1.f-wx),       w11 = wy*wx;
        float s0 = 0.f, s1 = 0.f, s2 = 0.f, s3 = 0.f;
        #pragma unroll
        for (int ti = 0; ti < 2; ++ti) {
            const int tl = ti ? tt1 : tt0;
            const float tw = ti ? al : (1.f - al);
            const int mb = (bb*WL + tl)*HH;
            const int i00 = (mb + y0i)*WW + x0i;
            const int i01 = (mb + y0i)*WW + x1i;
            const int i10 = (mb + y1i)*WW + x0i;
            const int i11 = (mb + y1i)*WW + x1i;
            const float k00 = msk[i00] ? 0.f : w00;
            const float k01 = msk[i01] ? 0.f : w01;
            const float k10 = msk[i10] ? 0.f : w10;
            const float k11 = msk[i11] ? 0.f : w11;
            float4 v00 = f4[(size_t)i00*64 + t];
            float4 v01 = f4[(size_t)i01*64 + t];
            float4 v10 = f4[(size_t)i10*64 + t];
            float4 v11 = f4[(size_t)i11*64 + t];
            s0 += tw*(k00*v00.x + k01*v01.x + k10*v10.x + k11*v11.x);
            s1 += tw*(k00*v00.y + k01*v01.y + k10*v10.y + k11*v11.y);
            s2 += tw*(k00*v00.z + k01*v01.z + k10*v10.z + k11*v11.z);
            s3 += tw*(k00*v00.w + k01*v01.w + k10*v10.w + k11*v11.w);
        }
        a0 += wgt*s0; a1 += wgt*s1; a2 += wgt*s2; a3 += wgt*s3;
    }
    float4 r; r.x = a0; r.y = a1; r.z = a2; r.w = a3;
    reinterpret_cast<float4*>(agg)[(size_t)bq*64 + t] = r;
}

// ---------------------------------------------------------------------------
// FP32 WMMA GEMM, templated on K/N/MODE so all strides are immediates.
// Each wave computes a 16x64 C tile: one A fragment feeds 4 WMMAs per k-step.
// Grid: (Mr/16, N/64), blockDim 32 (one wave).
// A [Mr,K] row-major, Bm [K,N] row-major.
// MODE: 1=+bias+residual(addsrc), 2=+bias+exact GELU, 3=+bias+accumulate into C.
// Fragment layouts per ISA 7.12.2 (A: lanes0-15 K={0,1}, lanes16-31 K={2,3};
// C: VGPR v -> rows v / v+8 for lane halves).
// ---------------------------------------------------------------------------
template<int K, int N, int MODE>
__global__ __launch_bounds__(32)
void gemm_wmma(const float* __restrict__ A, const float* __restrict__ Bm,
               const float* __restrict__ bias, const float* __restrict__ addsrc,
               float* __restrict__ C) {
    const int m0 = blockIdx.x * 16, n0 = blockIdx.y * 64;
    const int lane = threadIdx.x;
    const int half = lane >> 4, l = lane & 15;

    // Per-lane base pointers (single 64-bit setup; loop indexing is 32-bit).
    const float* Ap = A  + (m0 + l) * K + 2 * half;   // a.x=Ap[k], a.y=Ap[k+1]
    const float* Bp = Bm + (2 * half) * N + n0 + l;   // b.x=Bp[k*N+j*16], b.y=Bp[k*N+N+j*16]

    v8f acc0 = {}, acc1 = {}, acc2 = {}, acc3 = {};
    #pragma unroll 4
    for (int k = 0; k < K; k += 4) {
        v2f a;
        a.x = Ap[k];
        a.y = Ap[k + 1];
        // speculative prefetch of the B panel two k-steps ahead (non-faulting)
        __builtin_prefetch(Bp + (k + 8) * N, 0, 1);
        v2f b0, b1, b2, b3;
        b0.x = Bp[k*N +  0]; b0.y = Bp[k*N + N +  0];
        b1.x = Bp[k*N + 16]; b1.y = Bp[k*N + N + 16];
        b2.x = Bp[k*N + 32]; b2.y = Bp[k*N + N + 32];
        b3.x = Bp[k*N + 48]; b3.y = Bp[k*N + N + 48];
        acc0 = __builtin_amdgcn_wmma_f32_16x16x4_f32(false, a, false, b0, (short)0, acc0, false, false);
        acc1 = __builtin_amdgcn_wmma_f32_16x16x4_f32(false, a, false, b1, (short)0, acc1, false, false);
        acc2 = __builtin_amdgcn_wmma_f32_16x16x4_f32(false, a, false, b2, (short)0, acc2, false, false);
        acc3 = __builtin_amdgcn_wmma_f32_16x16x4_f32(false, a, false, b3, (short)0, acc3, false, false);
    }

    #pragma unroll
    for (int j = 0; j < 4; ++j) {
        const v8f acc = (j == 0) ? acc0 : (j == 1) ? acc1 : (j == 2) ? acc2 : acc3;
        const int col = n0 + j * 16 + l;
        const float bv = bias[col];
        #pragma unroll
        for (int v = 0; v < 8; ++v) {
            const int row = m0 + v + 8 * half;
            float x = acc[v] + bv;
            if (MODE == 1)      x += addsrc[row * N + col];
            else if (MODE == 2) x = 0.5f * x * (1.0f + erff(x * 0.70710678118654752f));
            else if (MODE == 3) x += C[row * N + col];
            C[row * N + col] = x;
        }
    }
}

// ---------------------------------------------------------------------------
extern "C" void kernel_launch(void* const* d_in, const int* in_sizes, int n_in,
                              void* d_out, int out_size, void* d_ws, size_t ws_size,
                              hipStream_t stream) {
    const float*         q         = (const float*)d_in[0];
    const float*         kv_feats  = (const float*)d_in[1];
    const unsigned char* kv_mask   = (const unsigned char*)d_in[2];
    const float*         delta_sec = (const float*)d_in[3];
    const float*         w_ref     = (const float*)d_in[4];
    const float*         b_ref     = (const float*)d_in[5];
    const float*         w_delta   = (const float*)d_in[6];
    const float*         b_delta   = (const float*)d_in[7];
    const float*         w_alpha   = (const float*)d_in[8];
    const float*         b_alpha   = (const float*)d_in[9];
    const float*         lam       = (const float*)d_in[10];
    const float*         w_proj    = (const float*)d_in[11];
    const float*         b_proj    = (const float*)d_in[12];
    const float*         ln1_g     = (const float*)d_in[13];
    const float*         ln1_b     = (const float*)d_in[14];
    const float*         ln2_g     = (const float*)d_in[15];
    const float*         ln2_b     = (const float*)d_in[16];
    const float*         w_ffn1    = (const float*)d_in[17];
    const float*         b_ffn1    = (const float*)d_in[18];
    const float*         w_ffn2    = (const float*)d_in[19];
    const float*         b_ffn2    = (const float*)d_in[20];
    const float*         w_rel     = (const float*)d_in[21];
    const float*         b_rel     = (const float*)d_in[22];

    float* out = (float*)d_out;
    float* ws  = (float*)d_ws;
    float* qn  = ws + WS_QN;
    float* rf  = ws + WS_REF;
    float* prm = ws + WS_PRM;
    float* agg = ws + WS_AGG;
    float* h   = ws + WS_H;
    float* mid = ws + WS_MID;

    ln_kernel<<<BQ, DD, 0, stream>>>(q, ln1_g, ln1_b, qn);
    ref_kernel<<<BQ, DD, 0, stream>>>(qn, w_ref, b_ref, rf);
    prep_kernel<<<BQ*WL, 64, 0, stream>>>(qn, delta_sec, w_rel, b_rel,
                                          w_delta, b_delta, w_alpha, b_alpha,
                                          rf, lam, prm);
    sample_kernel<<<BQ, 64, 0, stream>>>(kv_feats, kv_mask, prm, agg);

    // out = q + agg @ w_proj + b_proj
    gemm_wmma<DD, DD, 1><<<dim3(BQ/16, DD/64), 32, 0, stream>>>(agg, w_proj, b_proj, q, out);
    // h = LN2(out)
    ln_kernel<<<BQ, DD, 0, stream>>>(out, ln2_g, ln2_b, h);
    // mid = gelu(h @ w_ffn1 + b_ffn1)
    gemm_wmma<DD, 4*DD, 2><<<dim3(BQ/16, (4*DD)/64), 32, 0, stream>>>(h, w_ffn1, b_ffn1, nullptr, mid);
    // out += mid @ w_ffn2 + b_ffn2
    gemm_wmma<4*DD, DD, 3><<<dim3(BQ/16, DD/64), 32, 0, stream>>>(mid, w_ffn2, b_ffn2, nullptr, out);
}